// decoder_87222195847239
// MI455X (gfx1250) — compile-verified
//
#include <hip/hip_runtime.h>

// Problem constants (from reference)
static constexpr int kB   = 16;
static constexpr int kH   = 8;
static constexpr int kW   = 32;
static constexpr int kHW  = 256;      // H*W
static constexpr int kD   = 512;
static constexpr int kHID = 512;
static constexpr int kC   = 7000;
static constexpr int kT   = 40;
static constexpr int kSeq = 41;       // scan length (outputs drop step 0)

typedef __attribute__((ext_vector_type(2))) float v2f;
typedef __attribute__((ext_vector_type(8))) float v8f;

// ---------------------------------------------------------------------------
// f32 WMMA: D(16x16) = A(16x4) * B(4x16) + C   (one matrix per wave32)
// ---------------------------------------------------------------------------
__device__ __forceinline__ v8f wmma4(v2f a, v2f b, v8f c) {
  return __builtin_amdgcn_wmma_f32_16x16x4_f32(
      /*neg_a=*/false, a, /*neg_b=*/false, b,
      /*c_mod=*/(short)0, c, /*reuse_a=*/false, /*reuse_b=*/false);
}

// Lane helpers per ISA §7.12.2 (wave32):
//   A 16x4 f32: lane L<16 holds row m=L, k={0,1}; L>=16 holds m=L-16, k={2,3}
//   B  4x16:    mirrored (k split across lane halves, n = lane&15)
//   C/D 16x16:  acc[r] = D[m = r + (L>=16)*8][n = lane&15]

// ---------------------------------------------------------------------------
// misc small kernels
// ---------------------------------------------------------------------------
__global__ void zero_kernel(float* __restrict__ p, int n) {
  int i = blockIdx.x * blockDim.x + threadIdx.x;
  if (i < n) p[i] = 0.0f;
}

// xs[0] = hw ; xs[1][b][h] = w_lin1[h][C-3] + b_lin1[h]
__global__ void xs_init_kernel(const float* __restrict__ hw,
                               const float* __restrict__ w_lin1,
                               const float* __restrict__ b_lin1,
                               float* __restrict__ xs) {
  int i = blockIdx.x * blockDim.x + threadIdx.x;
  if (i >= kB * kHID) return;
  int h = i % kHID;
  xs[i] = hw[i];
  xs[kB * kHID + i] = w_lin1[(size_t)h * kC + (kC - 3)] + b_lin1[h];
}

// ---------------------------------------------------------------------------
// xs[t+2][b][h] = sum_c y[b][t][c] * w_lin1[h][c] + b_lin1[h]
// one wave per (t, group of 4 h-tiles): A fetch shared across 4 WMMAs
// ---------------------------------------------------------------------------
__global__ void embed_kernel(const float* __restrict__ y,
                             const float* __restrict__ w_lin1,
                             const float* __restrict__ b_lin1,
                             float* __restrict__ xs) {
  int gw   = (blockIdx.x * blockDim.x + threadIdx.x) >> 5;
  int lane = threadIdx.x & 31;
  const int ngroups = kHID / 64;                // 8
  if (gw >= (kT - 1) * ngroups) return;
  int t  = gw / ngroups;
  int n0 = (gw % ngroups) * 64;

  int am = lane & 15, ak = (lane >> 4) << 1;
  int bi = lane & 15, bk = (lane >> 4) << 1;

  const float* arow = y + (size_t)am * (kT - 1) * kC + (size_t)t * kC + ak;
  const float* brow = w_lin1 + (size_t)(n0 + bi) * kC + bk;

  v8f acc[4] = {{}, {}, {}, {}};
#pragma unroll 2
  for (int k0 = 0; k0 < kC; k0 += 4) {
    v2f a = *(const v2f*)(arow + k0);
#pragma unroll
    for (int tj = 0; tj < 4; ++tj) {
      v2f b = *(const v2f*)(brow + (size_t)tj * 16 * kC + k0);
      acc[tj] = wmma4(a, b, acc[tj]);
    }
  }
  int mb = (lane >> 4) * 8;
  float* dst = xs + (size_t)(t + 2) * kB * kHID;
#pragma unroll
  for (int tj = 0; tj < 4; ++tj) {
    int nn = n0 + tj * 16 + (lane & 15);
    float bias = b_lin1[nn];
#pragma unroll
    for (int r = 0; r < 8; ++r) dst[(mb + r) * kHID + nn] = acc[tj][r] + bias;
  }
}

// ---------------------------------------------------------------------------
// conv2V = conv3x3(V, w_conv2, SAME) + b_conv2   (done once)
// one wave per (b, h, group of 2 dout-tiles) covering both 16-wide w-halves:
//   2 spatial (M) tiles x 2 dout (N) tiles -> A shared over N, B shared over M
// ---------------------------------------------------------------------------
__global__ void conv2_kernel(const float* __restrict__ V,
                             const float* __restrict__ w2,
                             const float* __restrict__ b2,
                             float* __restrict__ out) {
  int gw = (blockIdx.x * blockDim.x + threadIdx.x) >> 5;
  if (gw >= kB * kH * (kD / 32)) return;        // 2048 waves
  int lane = threadIdx.x & 31;
  int ntg = gw & 15;
  int h   = (gw >> 4) & 7;
  int b   = gw >> 7;

  int am = lane & 15, ak = (lane >> 4) << 1;
  int bi = lane & 15, bk = (lane >> 4) << 1;
  int n0 = ntg * 32;

  const float* Vb = V + (size_t)b * kD * kHW;
  v8f acc00 = {}, acc01 = {}, acc10 = {}, acc11 = {};
  for (int dy = 0; dy < 3; ++dy) {
    int hh = h + dy - 1;
    if (hh < 0 || hh >= kH) continue;           // uniform over wave
    const float* vrow = Vb + hh * kW;
    for (int dx = 0; dx < 3; ++dx) {
      int ww0 = am + dx - 1;                    // w-half 0 spatial col
      int ww1 = 16 + am + dx - 1;               // w-half 1 spatial col
      bool v0 = (ww0 >= 0) && (ww0 < kW);
      bool v1 = (ww1 < kW);
      int o0 = v0 ? ww0 : 0;
      int o1 = v1 ? ww1 : 0;
      const float* wp0 = w2 + (size_t)(n0 + bi) * kD * 9 + dy * 3 + dx;
      const float* wp1 = wp0 + (size_t)16 * kD * 9;
#pragma unroll 4
      for (int k0 = 0; k0 < kD; k0 += 4) {
        int ka = k0 + ak, kb = k0 + bk;
        v2f a0, a1, b0, b1;
        a0.x = vrow[(size_t)ka * kHW + o0];
        a0.y = vrow[(size_t)(ka + 1) * kHW + o0];
        a1.x = vrow[(size_t)ka * kHW + o1];
        a1.y = vrow[(size_t)(ka + 1) * kHW + o1];
        if (!v0) { a0.x = 0.0f; a0.y = 0.0f; }
        if (!v1) { a1.x = 0.0f; a1.y = 0.0f; }
        b0.x = wp0[(size_t)kb * 9];
        b0.y = wp0[(size_t)(kb + 1) * 9];
        b1.x = wp1[(size_t)kb * 9];
        b1.y = wp1[(size_t)(kb + 1) * 9];
        acc00 = wmma4(a0, b0, acc00);
        acc01 = wmma4(a0, b1, acc01);
        acc10 = wmma4(a1, b0, acc10);
        acc11 = wmma4(a1, b1, acc11);
      }
    }
  }
  int mb = (lane >> 4) * 8;
  float* ob = out + (size_t)b * kD * kHW;
#pragma unroll
  for (int r = 0; r < 8; ++r) {
    int nn0 = n0 + (lane & 15), nn1 = nn0 + 16;
    float bi0 = b2[nn0], bi1 = b2[nn1];
    int w0 = mb + r, w1 = 16 + mb + r;
    ob[(size_t)nn0 * kHW + h * kW + w0] = acc00[r] + bi0;
    ob[(size_t)nn1 * kHW + h * kW + w0] = acc01[r] + bi1;
    ob[(size_t)nn0 * kHW + h * kW + w1] = acc10[r] + bi0;
    ob[(size_t)nn1 * kHW + h * kW + w1] = acc11[r] + bi1;
  }
}

// ---------------------------------------------------------------------------
// gates[b][n] = b_ih[n]+b_hh[n] + x@W_ih^T + h@W_hh^T    N=2048, K=512 (x2)
// one wave per group of 4 n-tiles (64 cols): A fetch shared across 4 WMMAs
// ---------------------------------------------------------------------------
__global__ void lstm_gates_kernel(const float* __restrict__ x,
                                  const float* __restrict__ h,
                                  const float* __restrict__ Wih,
                                  const float* __restrict__ bih,
                                  const float* __restrict__ Whh,
                                  const float* __restrict__ bhh,
                                  float* __restrict__ gates) {
  int gw = (blockIdx.x * blockDim.x + threadIdx.x) >> 5;
  if (gw >= (4 * kHID) / 64) return;            // 32 waves
  int lane = threadIdx.x & 31;
  int n0 = gw * 64;
  int am = lane & 15, ak = (lane >> 4) << 1;
  int bi = lane & 15, bk = (lane >> 4) << 1;

  const float* xr = x + am * kHID + ak;
  const float* hr = h + am * kHID + ak;
  const float* wi = Wih + (size_t)(n0 + bi) * kHID + bk;
  const float* wh = Whh + (size_t)(n0 + bi) * kHID + bk;

  v8f acc[4] = {{}, {}, {}, {}};
#pragma unroll 4
  for (int k0 = 0; k0 < kHID; k0 += 4) {
    v2f a = *(const v2f*)(xr + k0);
#pragma unroll
    for (int tj = 0; tj < 4; ++tj) {
      v2f b = *(const v2f*)(wi + (size_t)tj * 16 * kHID + k0);
      acc[tj] = wmma4(a, b, acc[tj]);
    }
  }
#pragma unroll 4
  for (int k0 = 0; k0 < kHID; k0 += 4) {
    v2f a = *(const v2f*)(hr + k0);
#pragma unroll
    for (int tj = 0; tj < 4; ++tj) {
      v2f b = *(const v2f*)(wh + (size_t)tj * 16 * kHID + k0);
      acc[tj] = wmma4(a, b, acc[tj]);
    }
  }
  int mb = (lane >> 4) * 8;
#pragma unroll
  for (int tj = 0; tj < 4; ++tj) {
    int nn = n0 + tj * 16 + (lane & 15);
    float bias = bih[nn] + bhh[nn];
#pragma unroll
    for (int r = 0; r < 8; ++r) gates[(mb + r) * (4 * kHID) + nn] = acc[tj][r] + bias;
  }
}

__global__ void lstm_update_kernel(const float* __restrict__ g,
                                   float* __restrict__ h, float* __restrict__ c) {
  int idx = blockIdx.x * blockDim.x + threadIdx.x;
  if (idx >= kB * kHID) return;
  int b = idx / kHID, j = idx % kHID;
  const float* gb = g + b * 4 * kHID;
  float gi = gb[j], gf = gb[kHID + j], gg = gb[2 * kHID + j], go = gb[3 * kHID + j];
  float si = 1.0f / (1.0f + __expf(-gi));
  float sf = 1.0f / (1.0f + __expf(-gf));
  float so = 1.0f / (1.0f + __expf(-go));
  float c2 = sf * c[idx] + si * tanhf(gg);
  c[idx] = c2;
  h[idx] = so * tanhf(c2);
}

// ---------------------------------------------------------------------------
// generic M=16 GEMM: out[m][n] = bias[n] + sum_k A[m][k]*Bm[n][k]
// one wave per group of 4 n-tiles; A row stride = lda; B is (N,K) row-major
// ---------------------------------------------------------------------------
__global__ void gemm16_kernel(const float* __restrict__ A, int lda,
                              const float* __restrict__ Bm,
                              const float* __restrict__ bias,
                              float* __restrict__ out, int N, int K) {
  int gw = (blockIdx.x * blockDim.x + threadIdx.x) >> 5;
  int ngroups = (N + 63) >> 6;
  if (gw >= ngroups) return;
  int lane = threadIdx.x & 31;
  int n0 = gw << 6;
  int am = lane & 15, ak = (lane >> 4) << 1;
  int bi = lane & 15, bk = (lane >> 4) << 1;

  const float* arow = A + (size_t)am * lda + ak;
  const float* brow[4];
  bool bv[4];
#pragma unroll
  for (int tj = 0; tj < 4; ++tj) {
    int nn = n0 + tj * 16 + bi;
    bv[tj] = nn < N;
    brow[tj] = Bm + (size_t)(bv[tj] ? nn : (N - 1)) * K + bk;
  }

  v8f acc[4] = {{}, {}, {}, {}};
#pragma unroll 4
  for (int k0 = 0; k0 < K; k0 += 4) {
    v2f a = *(const v2f*)(arow + k0);
#pragma unroll
    for (int tj = 0; tj < 4; ++tj) {
      v2f b = *(const v2f*)(brow[tj] + k0);
      if (!bv[tj]) { b.x = 0.0f; b.y = 0.0f; }
      acc[tj] = wmma4(a, b, acc[tj]);
    }
  }
  int mb = (lane >> 4) * 8;
#pragma unroll
  for (int tj = 0; tj < 4; ++tj) {
    if (bv[tj]) {
      int nn = n0 + tj * 16 + bi;
      float bb = bias[nn];
#pragma unroll
      for (int r = 0; r < 8; ++r) out[(size_t)(mb + r) * N + nn] = acc[tj][r] + bb;
    }
  }
}

// ---------------------------------------------------------------------------
// attention: logits -> softmax over 256 spatial -> glimpse; writes attw to out
// one block per batch row, 256 threads = one per spatial position
// ---------------------------------------------------------------------------
__global__ void attention_kernel(const float* __restrict__ conv2V,
                                 const float* __restrict__ hproj,
                                 const float* __restrict__ w3,
                                 const float* __restrict__ b3,
                                 const float* __restrict__ Vin,
                                 float* __restrict__ glimpse,
                                 float* __restrict__ att_out) {
  __shared__ float red[256];
  __shared__ float aw_sm[256];
  int b = blockIdx.x, pos = threadIdx.x;

  const float* cv = conv2V + (size_t)b * kD * kHW + pos;
  const float* hp = hproj + (size_t)b * kD;
  float logit = b3[0];
#pragma unroll 4
  for (int d = 0; d < kD; ++d)
    logit += tanhf(cv[(size_t)d * kHW] + hp[d]) * w3[d];

  // softmax over 256 positions
  red[pos] = logit; __syncthreads();
  for (int s = 128; s > 0; s >>= 1) {
    if (pos < s) red[pos] = fmaxf(red[pos], red[pos + s]);
    __syncthreads();
  }
  float mx = red[0]; __syncthreads();
  float e = __expf(logit - mx);
  red[pos] = e; __syncthreads();
  for (int s = 128; s > 0; s >>= 1) {
    if (pos < s) red[pos] += red[pos + s];
    __syncthreads();
  }
  float aw = e / red[0];
  aw_sm[pos] = aw;
  att_out[(size_t)b * kT * kHW + pos] = aw;
  __syncthreads();

  // glimpse[b][d] = sum_pos V[b][d][pos] * aw[pos]
  const float* Vb = Vin + (size_t)b * kD * kHW;
  for (int d = pos; d < kD; d += 256) {
    float s = 0.0f;
#pragma unroll 4
    for (int p = 0; p < kHW; ++p) s += Vb[(size_t)d * kHW + p] * aw_sm[p];
    glimpse[(size_t)b * kD + d] = s;
  }
}

__global__ void concat_kernel(const float* __restrict__ h2,
                              const float* __restrict__ gl,
                              float* __restrict__ hg) {
  int idx = blockIdx.x * blockDim.x + threadIdx.x;
  if (idx >= kB * (kHID + kD)) return;
  int b = idx / (kHID + kD), k = idx % (kHID + kD);
  hg[idx] = (k < kHID) ? h2[b * kHID + k] : gl[b * kD + (k - kHID)];
}

__global__ void logsoftmax_kernel(const float* __restrict__ z,
                                  float* __restrict__ out) {
  __shared__ float red[256];
  int b = blockIdx.x, tid = threadIdx.x;
  const float* zb = z + (size_t)b * kC;
  float mx = -3.402823466e38f;
  for (int n = tid; n < kC; n += 256) mx = fmaxf(mx, zb[n]);
  red[tid] = mx; __syncthreads();
  for (int s = 128; s > 0; s >>= 1) {
    if (tid < s) red[tid] = fmaxf(red[tid], red[tid + s]);
    __syncthreads();
  }
  mx = red[0]; __syncthreads();
  float sum = 0.0f;
  for (int n = tid; n < kC; n += 256) sum += __expf(zb[n] - mx);
  red[tid] = sum; __syncthreads();
  for (int s = 128; s > 0; s >>= 1) {
    if (tid < s) red[tid] += red[tid + s];
    __syncthreads();
  }
  float lse = mx + __logf(red[0]);
  float* ob = out + (size_t)b * kT * kC;
  for (int n = tid; n < kC; n += 256) ob[n] = zb[n] - lse;
}

// ---------------------------------------------------------------------------
extern "C" void kernel_launch(void* const* d_in, const int* in_sizes, int n_in,
                              void* d_out, int out_size, void* d_ws, size_t ws_size,
                              hipStream_t stream) {
  (void)in_sizes; (void)n_in; (void)out_size; (void)ws_size;
  const float* hw      = (const float*)d_in[0];
  const float* y       = (const float*)d_in[1];
  const float* V       = (const float*)d_in[2];
  const float* w_lin1  = (const float*)d_in[3];
  const float* b_lin1  = (const float*)d_in[4];
  const float* W_ih    = (const float*)d_in[5];
  const float* b_ih    = (const float*)d_in[6];
  const float* W_hh    = (const float*)d_in[7];
  const float* b_hh    = (const float*)d_in[8];
  const float* w_conv1 = (const float*)d_in[9];
  const float* b_conv1 = (const float*)d_in[10];
  const float* w_conv2 = (const float*)d_in[11];
  const float* b_conv2 = (const float*)d_in[12];
  const float* w_conv3 = (const float*)d_in[13];
  const float* b_conv3 = (const float*)d_in[14];
  const float* w_lin2  = (const float*)d_in[15];
  const float* b_lin2  = (const float*)d_in[16];

  float* outF   = (float*)d_out;
  float* outAtt = outF + (size_t)kB * kT * kC;

  // workspace layout (~10.6 MB)
  float* ws = (float*)d_ws;
  size_t off = 0;
  float* conv2V = ws + off; off += (size_t)kB * kD * kHW;    // 2,097,152
  float* xs     = ws + off; off += (size_t)kSeq * kB * kHID; //   335,872
  float* h1     = ws + off; off += kB * kHID;
  float* c1     = ws + off; off += kB * kHID;
  float* h2     = ws + off; off += kB * kHID;
  float* c2     = ws + off; off += kB * kHID;
  float* gates  = ws + off; off += kB * 4 * kHID;
  float* hproj  = ws + off; off += kB * kD;
  float* glimpse= ws + off; off += kB * kD;
  float* hg     = ws + off; off += kB * (kHID + kD);
  float* zbuf   = ws + off; off += (size_t)kB * kC;

  // init state (h1,c1,h2,c2 contiguous)
  zero_kernel<<<(4 * kB * kHID + 255) / 256, 256, 0, stream>>>(h1, 4 * kB * kHID);
  xs_init_kernel<<<(kB * kHID + 255) / 256, 256, 0, stream>>>(hw, w_lin1, b_lin1, xs);
  // 39 t * 8 groups = 312 waves, 8 waves/block
  embed_kernel<<<39, 256, 0, stream>>>(y, w_lin1, b_lin1, xs);
  // 2048 waves
  conv2_kernel<<<256, 256, 0, stream>>>(V, w_conv2, b_conv2, conv2V);

  for (int t = 0; t < kSeq; ++t) {
    lstm_gates_kernel<<<4, 256, 0, stream>>>(xs + (size_t)t * kB * kHID, h1,
                                             W_ih, b_ih, W_hh, b_hh, gates);
    lstm_update_kernel<<<32, 256, 0, stream>>>(gates, h1, c1);
    lstm_gates_kernel<<<4, 256, 0, stream>>>(h1, h2, W_ih, b_ih, W_hh, b_hh, gates);
    lstm_update_kernel<<<32, 256, 0, stream>>>(gates, h2, c2);
    if (t == 0) continue;  // step-0 outputs are discarded by the reference

    // hproj: N=512 -> 8 groups (1 block of 8 waves)
    gemm16_kernel<<<1, 256, 0, stream>>>(h2, kHID, w_conv1, b_conv1, hproj, kD, kHID);
    attention_kernel<<<kB, 256, 0, stream>>>(conv2V, hproj, w_conv3, b_conv3, V,
                                             glimpse, outAtt + (size_t)(t - 1) * kHW);
    concat_kernel<<<(kB * (kHID + kD) + 255) / 256, 256, 0, stream>>>(h2, glimpse, hg);
    // N=7000 -> 110 groups -> 14 blocks * 8 waves
    gemm16_kernel<<<14, 256, 0, stream>>>(hg, kHID + kD, w_lin2, b_lin2, zbuf,
                                          kC, kHID + kD);
    logsoftmax_kernel<<<kB, 256, 0, stream>>>(zbuf, outF + (size_t)(t - 1) * kC);
  }
}